// CopyGenerator_13855564497442
// MI455X (gfx1250) — compile-verified
//
#include <hip/hip_runtime.h>
#include <math.h>

typedef __attribute__((ext_vector_type(2))) float v2f;
typedef __attribute__((ext_vector_type(8))) float v8f;

#define TLEN    50
#define BATCH   16
#define DIN     512
#define VOCAB   20000
#define CVOCAB  20400
#define SRCLEN  200

#define NT 80      // vocab columns per block (5 waves x 16)
#define KC 64      // K chunk staged in LDS
#define AS 516     // A LDS row stride (floats): %4==0 for float4, bank-advance 4 -> conflict-free
#define BS 68      // B LDS row stride (floats): same properties

// ---------------------------------------------------------------------------
// Kernel A: p_copy[t*16+b] = sigmoid(hidden[t,b,:] . Wc + bc)   (800 rows)
// ---------------------------------------------------------------------------
__global__ __launch_bounds__(256)
void pcopy_kernel(const float* __restrict__ hidden, const float* __restrict__ Wc,
                  const float* __restrict__ bc, float* __restrict__ pcopy) {
    int row  = blockIdx.x * 8 + (threadIdx.x >> 5);
    if (row >= TLEN * BATCH) return;
    int lane = threadIdx.x & 31;
    float s = 0.f;
    #pragma unroll 4
    for (int k = lane; k < DIN; k += 32) s += hidden[(size_t)row * DIN + k] * Wc[k];
    #pragma unroll
    for (int off = 16; off > 0; off >>= 1) s += __shfl_down(s, off);
    if (lane == 0) pcopy[row] = 1.0f / (1.0f + expf(-(s + bc[0])));
}

// ---------------------------------------------------------------------------
// Kernel C: recover one-hot indices: idx[s*16+b] = v with src_map[s,b,v] != 0
// Streams the dense 261 MB one-hot exactly once (the roofline-dominant pass).
// ---------------------------------------------------------------------------
__global__ __launch_bounds__(256)
void onehot_idx_kernel(const float* __restrict__ src_map, int* __restrict__ idx) {
    int row = blockIdx.x;                       // s*16+b, 3200 rows
    const float4* p = (const float4*)(src_map + (size_t)row * CVOCAB);
    for (int i = threadIdx.x; i < CVOCAB / 4; i += 256) {
        float4 v = p[i];
        if (v.x != 0.f) idx[row] = 4 * i + 0;
        if (v.y != 0.f) idx[row] = 4 * i + 1;
        if (v.z != 0.f) idx[row] = 4 * i + 2;
        if (v.w != 0.f) idx[row] = 4 * i + 3;
    }
}

// ---------------------------------------------------------------------------
// Zero the extended-vocab tail columns [20000, 20400)
// ---------------------------------------------------------------------------
__global__ __launch_bounds__(256)
void zero_tail_kernel(float* __restrict__ out) {
    int i = blockIdx.x * 256 + threadIdx.x;
    if (i < TLEN * BATCH * (CVOCAB - VOCAB)) {
        int tb = i / (CVOCAB - VOCAB);
        int r  = i % (CVOCAB - VOCAB);
        out[(size_t)tb * CVOCAB + VOCAB + r] = 0.f;
    }
}

// ---------------------------------------------------------------------------
// Kernel B: fused  logits = hidden@W^T + b ; mask batch==1 ; softmax over the
// 16 batches (rows of each WMMA tile) ; * (1-p_copy) ; store to out[:, :, :V].
// Block = 5 waves (160 thr): one t (M=16 = all batches), 80 vocab columns.
// Uses V_WMMA_F32_16X16X4_F32 (exact fp32, K=4 per instruction).
// ---------------------------------------------------------------------------
__global__ __launch_bounds__(160)
void gemm_softmax_kernel(const float* __restrict__ hidden, const float* __restrict__ W,
                         const float* __restrict__ bvec, const float* __restrict__ pcopy,
                         float* __restrict__ out) {
    __shared__ float Alds[16 * AS];     // 33.0 KB  (hidden rows for this t, full K)
    __shared__ float Blds[NT * BS];     // 21.8 KB  (W tile, one K-chunk)
    __shared__ float pc_s[16];

    const int t     = blockIdx.y;
    const int nbase = blockIdx.x * NT;
    const int tid   = threadIdx.x;

    if (tid < 16) pc_s[tid] = pcopy[t * 16 + tid];

    // Stage A: 16 x 512 fp32 (row m = batch), float4-coalesced
    const float4* hsrc = (const float4*)(hidden + (size_t)t * 16 * DIN);
    for (int i = tid; i < 16 * (DIN / 4); i += 160) {
        int row = i >> 7, j4 = i & 127;
        *(float4*)&Alds[row * AS + j4 * 4] = hsrc[row * 128 + j4];
    }

    const int wave = tid >> 5;
    const int lane = tid & 31;
    const int lm   = lane & 15;          // A: M row / B: N col / C: N col
    const int koff = (lane >> 4) << 1;   // +2 K-offset for upper half-wave
    const int half = lane >> 4;

    v8f c = {0.f, 0.f, 0.f, 0.f, 0.f, 0.f, 0.f, 0.f};

    for (int kc = 0; kc < DIN; kc += KC) {
        __syncthreads();
        // Stage B chunk: 80 rows (vocab) x 64 K, float4-coalesced
        for (int i = tid; i < NT * (KC / 4); i += 160) {
            int row = i >> 4, j4 = i & 15;
            *(float4*)&Blds[row * BS + j4 * 4] =
                *(const float4*)&W[(size_t)(nbase + row) * DIN + kc + j4 * 4];
        }
        __syncthreads();
        #pragma unroll
        for (int k0 = 0; k0 < KC; k0 += 4) {
            v2f a  = *(const v2f*)&Alds[lm * AS + kc + k0 + koff];
            v2f bb = *(const v2f*)&Blds[(wave * 16 + lm) * BS + k0 + koff];
            c = __builtin_amdgcn_wmma_f32_16x16x4_f32(false, a, false, bb,
                                                      (short)0, c, false, false);
        }
    }

    // ---- epilogue: bias, -inf mask at batch 1, softmax over 16 rows, scale ----
    const int   n    = nbase + wave * 16 + lm;
    const float bias = bvec[n];
    #pragma unroll
    for (int j = 0; j < 8; ++j) c[j] += bias;
    if (half == 0) c[1] = -__builtin_inff();          // row m=1 (VGPR1, lanes 0-15)

    float mx = c[0];
    #pragma unroll
    for (int j = 1; j < 8; ++j) mx = fmaxf(mx, c[j]);
    mx = fmaxf(mx, __shfl_xor(mx, 16));               // combine rows 0-7 with 8-15

    float e[8], s = 0.f;
    #pragma unroll
    for (int j = 0; j < 8; ++j) { e[j] = expf(c[j] - mx); s += e[j]; }
    s += __shfl_xor(s, 16);
    const float inv = 1.0f / s;

    #pragma unroll
    for (int j = 0; j < 8; ++j) {
        int m = j + half * 8;                          // batch index
        out[(size_t)(t * 16 + m) * CVOCAB + n] = e[j] * inv * (1.0f - pc_s[m]);
    }
}

// ---------------------------------------------------------------------------
// Kernel D: permuted copy-scatter.
// final[t,b,v] += sum_s attn[ts,bs,s]*p_copy[ts,bs] where ts,bs = divmod(b*50+t,16)
// ---------------------------------------------------------------------------
__global__ __launch_bounds__(256)
void scatter_kernel(const float* __restrict__ attn, const float* __restrict__ pcopy,
                    const int* __restrict__ idx, float* __restrict__ out) {
    int pair = blockIdx.x * 8 + (threadIdx.x >> 5);   // out linear (t*16+b)
    if (pair >= TLEN * BATCH) return;
    int lane = threadIdx.x & 31;
    int t = pair >> 4, b = pair & 15;
    int src = b * TLEN + t;
    int ts = src >> 4, bs = src & 15;
    float scale = pcopy[ts * 16 + bs];
    const float* arow = attn + (size_t)(ts * 16 + bs) * SRCLEN;
    float* orow = out + (size_t)pair * CVOCAB;
    for (int s = lane; s < SRCLEN; s += 32)
        atomicAdd(&orow[idx[s * 16 + bs]], arow[s] * scale);
}

// ---------------------------------------------------------------------------
extern "C" void kernel_launch(void* const* d_in, const int* in_sizes, int n_in,
                              void* d_out, int out_size, void* d_ws, size_t ws_size,
                              hipStream_t stream) {
    const float* hidden  = (const float*)d_in[0];
    const float* attn    = (const float*)d_in[1];
    const float* src_map = (const float*)d_in[2];
    const float* W       = (const float*)d_in[3];
    const float* bvec    = (const float*)d_in[4];
    const float* Wc      = (const float*)d_in[5];
    const float* bc      = (const float*)d_in[6];
    float* out = (float*)d_out;

    float* pcopy = (float*)d_ws;                                   // 800 floats
    int*   idx   = (int*)((char*)d_ws + 1024 * sizeof(float));     // 3200 ints

    pcopy_kernel<<<100, 256, 0, stream>>>(hidden, Wc, bc, pcopy);
    onehot_idx_kernel<<<SRCLEN * BATCH, 256, 0, stream>>>(src_map, idx);
    zero_tail_kernel<<<(TLEN * BATCH * (CVOCAB - VOCAB) + 255) / 256, 256, 0, stream>>>(out);

    dim3 grid(VOCAB / NT, TLEN);   // (250, 50)
    gemm_softmax_kernel<<<grid, 160, 0, stream>>>(hidden, W, bvec, pcopy, out);

    scatter_kernel<<<100, 256, 0, stream>>>(attn, pcopy, idx, out);
}